// RecursiveLogit_19945828122903
// MI455X (gfx1250) — compile-verified
//
#include <hip/hip_runtime.h>
#include <hip/hip_bf16.h>
#include <math.h>

typedef float v2f __attribute__((ext_vector_type(2)));
typedef float v8f __attribute__((ext_vector_type(8)));

#define WAVES_PER_BLOCK 8
#define TBLOCK 256

// One wave computes util/eu for a 16-edge tile:
//   C(16x16) = A(16x32 rows of feats) x B(32x16), B's every column = W.
// Chained as 8x V_WMMA_F32_16X16X4_F32 (full fp32 precision).
// A layout (ISA 7.12.2, 32-bit A 16x4): lane<16 holds K,K+1 ; lane>=16 holds K+2,K+3.
// C layout: VGPR r, lanes 0-15 -> M=r ; lanes 16-31 -> M=r+8. Lanes 0 and 16 store.
__global__ __launch_bounds__(TBLOCK) void util_eu_kernel(
    const float* __restrict__ feats, const float* __restrict__ W,
    const float* __restrict__ bptr, float* __restrict__ util_out,
    float* __restrict__ eu_out, int n_edges)
{
    const int lane = threadIdx.x & 31;
    const int wave = threadIdx.x >> 5;
    const long tile = (long)blockIdx.x * WAVES_PER_BLOCK + wave;
    const long base = tile * 16;
    if (base >= n_edges) return;                 // wave-uniform
    int row = (int)base + (lane & 15);
    if (row >= n_edges) row = n_edges - 1;       // tail clamp (E%128==0 -> never taken)
    const int khalf = (lane >> 4) << 1;          // 0 or 2
    const float bias = bptr[0];

    v8f c = {0.f, 0.f, 0.f, 0.f, 0.f, 0.f, 0.f, 0.f};
#pragma unroll
    for (int s = 0; s < 8; ++s) {
        const int col = (s << 2) + khalf;
        const float2 av = *(const float2*)(feats + (size_t)row * 32 + col);
        v2f a;  a.x = av.x;   a.y = av.y;
        v2f bm; bm.x = W[col]; bm.y = W[col + 1];   // every B column = W
        c = __builtin_amdgcn_wmma_f32_16x16x4_f32(false, a, false, bm,
                                                  (short)0, c, false, false);
    }

    float u[8], e[8];
#pragma unroll
    for (int r = 0; r < 8; ++r) {
        float x = c[r] + bias;
        x = fminf(fmaxf(x, -100.0f), -1e-6f);
        u[r] = x;
        e[r] = expf(x);
    }
    if ((lane & 15) == 0) {                      // lane 0 -> M 0..7, lane 16 -> M 8..15
        const long off = base + (long)((lane >> 4) << 3);
        if (off + 8 <= n_edges) {
            *(float4*)(util_out + off)     = make_float4(u[0], u[1], u[2], u[3]);
            *(float4*)(util_out + off + 4) = make_float4(u[4], u[5], u[6], u[7]);
            *(float4*)(eu_out   + off)     = make_float4(e[0], e[1], e[2], e[3]);
            *(float4*)(eu_out   + off + 4) = make_float4(e[4], e[5], e[6], e[7]);
        }
    }
}

__global__ __launch_bounds__(TBLOCK) void init_z_kernel(
    const unsigned char* __restrict__ mask, float* __restrict__ z, int n)
{
    int i = blockIdx.x * blockDim.x + threadIdx.x;
    if (i < n) z[i] = mask[i] ? 1.0f : 0.0f;
}

__global__ __launch_bounds__(TBLOCK) void scatter_kernel(
    const int* __restrict__ src, const int* __restrict__ dst,
    const float* __restrict__ eu, const float* __restrict__ z_cur,
    float* __restrict__ z_next, int n_edges)
{
    int i = blockIdx.x * blockDim.x + threadIdx.x;
    if (i < n_edges) {
        float v = eu[i] * z_cur[dst[i]];         // gather: z fits in L2
        atomicAdd(&z_next[src[i]], v);           // global_atomic_add_f32 (no return)
    }
}

__global__ __launch_bounds__(TBLOCK) void value_kernel(
    const float* __restrict__ z, float* __restrict__ value_out, int n)
{
    int i = blockIdx.x * blockDim.x + threadIdx.x;
    if (i < n) value_out[i] = logf(z[i]);
}

__global__ __launch_bounds__(TBLOCK) void prob_kernel(
    const int* __restrict__ src, const int* __restrict__ dst,
    const float* __restrict__ eu, const float* __restrict__ z,
    float* __restrict__ prob_out, int n_edges)
{
    int i = blockIdx.x * blockDim.x + threadIdx.x;
    if (i < n_edges) prob_out[i] = eu[i] * z[dst[i]] / z[src[i]];
}

extern "C" void kernel_launch(void* const* d_in, const int* in_sizes, int n_in,
                              void* d_out, int out_size, void* d_ws, size_t ws_size,
                              hipStream_t stream) {
    const float*         feats = (const float*)d_in[0];
    const unsigned char* dmask = (const unsigned char*)d_in[1]; // jnp bool = 1 byte
    const int*           eidx  = (const int*)d_in[2];
    const float*         W     = (const float*)d_in[4];
    const float*         b     = (const float*)d_in[5];

    const int n_nodes = in_sizes[1];
    const int n_edges = in_sizes[2] / 2;
    const int* src = eidx;
    const int* dst = eidx + n_edges;

    // workspace: eu[n_edges] | zA[n_nodes] | zB[n_nodes]
    float* eu = (float*)d_ws;
    float* zA = eu + n_edges;
    float* zB = zA + n_nodes;

    float* out_value = (float*)d_out;
    float* out_util  = out_value + n_nodes;
    float* out_prob  = out_util + n_edges;

    const int tiles   = (n_edges + 15) / 16;
    const int ublocks = (tiles + WAVES_PER_BLOCK - 1) / WAVES_PER_BLOCK;
    const int nb_node = (n_nodes + TBLOCK - 1) / TBLOCK;
    const int nb_edge = (n_edges + TBLOCK - 1) / TBLOCK;

    // util + eu via f32 WMMA
    util_eu_kernel<<<ublocks, TBLOCK, 0, stream>>>(feats, W, b, out_util, eu, n_edges);

    // z0 = bvec ; 64x: z_next = scatter(eu * z[dst] -> src) + bvec
    init_z_kernel<<<nb_node, TBLOCK, 0, stream>>>(dmask, zA, n_nodes);
    float* zc = zA;
    float* zn = zB;
    for (int it = 0; it < 64; ++it) {
        init_z_kernel<<<nb_node, TBLOCK, 0, stream>>>(dmask, zn, n_nodes);
        scatter_kernel<<<nb_edge, TBLOCK, 0, stream>>>(src, dst, eu, zc, zn, n_edges);
        float* t = zc; zc = zn; zn = t;
    }

    value_kernel<<<nb_node, TBLOCK, 0, stream>>>(zc, out_value, n_nodes);
    prob_kernel<<<nb_edge, TBLOCK, 0, stream>>>(src, dst, eu, zc, out_prob, n_edges);
}